// GCNModelWithRegularization_79963701117031
// MI455X (gfx1250) — compile-verified
//
#include <hip/hip_runtime.h>
#include <hip/hip_bf16.h>
#include <math.h>

// ---------------------------------------------------------------------------
// GraphConv x2 + log_softmax for MI455X (gfx1250, wave32).
// Dense parts use V_WMMA_F32_16X16X4_F32 (full fp32 precision: the workload is
// scatter/atomic bound, GEMM flops are ~2.5 GF total, so no reason to quantize).
// ---------------------------------------------------------------------------

typedef __attribute__((ext_vector_type(2))) float v2f;
typedef __attribute__((ext_vector_type(8))) float v8f;

#define N_NODES 50000
#define N_EDGES 800000
#define IN_CH   64
#define HID_CH  128
#define OUT_CH  40

// ---------------------------------------------------------------------------
__global__ void zero_f4(float4* __restrict__ p, int n4) {
  int i = blockIdx.x * blockDim.x + threadIdx.x;
  if (i < n4) p[i] = make_float4(0.f, 0.f, 0.f, 0.f);
}

// ---------------------------------------------------------------------------
// segment_sum(feat[src], dst): (1<<LOG2G) lanes per edge, 4 channels per lane.
template <int LOG2G>
__global__ void scatter_add_kernel(const float* __restrict__ feat,
                                   const int* __restrict__ srcv,
                                   const int* __restrict__ dstv,
                                   float* __restrict__ agg) {
  const int groups = 1 << LOG2G;  // channel groups of 4
  const int ch = groups * 4;
  long long idx = (long long)blockIdx.x * blockDim.x + threadIdx.x;
  const long long total = (long long)N_EDGES << LOG2G;
  if (idx >= total) return;
  const int e = (int)(idx >> LOG2G);
  const int g = (int)(idx & (groups - 1));
  const int s = srcv[e];
  const int d = dstv[e];
  const float4 v = *(const float4*)(feat + (long long)s * ch + g * 4);
  float* o = agg + (long long)d * ch + g * 4;
  atomicAdd(o + 0, v.x);
  atomicAdd(o + 1, v.y);
  atomicAdd(o + 2, v.z);
  atomicAdd(o + 3, v.w);
}

// ---------------------------------------------------------------------------
// Fragment loaders for V_WMMA_F32_16X16X4_F32.
// A 16x4 (MxK): lanes 0-15 hold M=lane, {VGPR0,VGPR1} = {K0,K1};
//               lanes 16-31 hold M=lane-16, {K2,K3}.  (ISA 7.12.2)
// B 4x16 (KxN) assumed mirrored with N on lanes.
__device__ __forceinline__ v2f load_frag(const float* base) {
  return *(const v2f*)base;  // base already points at [row][k + khalf]
}

// ---------------------------------------------------------------------------
// h = relu(agg1 @ W1_rel^T + x @ W1_root^T + b1).  M=50000, K=64, N=128.
// One wave per 16x16 tile; 8 waves/block cover all 128 output channels.
__global__ __launch_bounds__(256) void gemm1_relu(
    const float* __restrict__ agg1, const float* __restrict__ x,
    const float* __restrict__ Wrel, const float* __restrict__ Wroot,
    const float* __restrict__ bias, float* __restrict__ h) {
  const int lane = threadIdx.x & 31;
  const int wave = threadIdx.x >> 5;
  const int m_base = blockIdx.x * 16;      // 3125 blocks, exact
  const int n_base = wave * 16;            // 8 waves * 16 = 128
  const int l15 = lane & 15;
  const int kh = (lane >> 4) * 2;          // 0 or 2 (K half select)
  const int arow = m_base + l15;
  const int bcol = n_base + l15;

  const float* A0 = agg1 + (long long)arow * IN_CH + kh;
  const float* X0 = x + (long long)arow * IN_CH + kh;
  const float* Wr = Wrel + (long long)bcol * IN_CH + kh;   // [out,in] row-major
  const float* Wo = Wroot + (long long)bcol * IN_CH + kh;

  v8f c = {};
#pragma unroll
  for (int k = 0; k < IN_CH; k += 4) {
    v2f a = load_frag(A0 + k);
    v2f b = load_frag(Wr + k);
    c = __builtin_amdgcn_wmma_f32_16x16x4_f32(false, a, false, b, (short)0, c,
                                              false, false);
  }
#pragma unroll
  for (int k = 0; k < IN_CH; k += 4) {
    v2f a = load_frag(X0 + k);
    v2f b = load_frag(Wo + k);
    c = __builtin_amdgcn_wmma_f32_16x16x4_f32(false, a, false, b, (short)0, c,
                                              false, false);
  }

  const float bn = bias[bcol];
  const int rbase = m_base + ((lane >> 4) << 3);  // C/D: lanes 16-31 hold M=8+i
#pragma unroll
  for (int i = 0; i < 8; ++i) {
    float v = c[i] + bn;
    h[(long long)(rbase + i) * HID_CH + bcol] = fmaxf(v, 0.0f);
  }
}

// ---------------------------------------------------------------------------
// out = log_softmax(agg2 @ W2_rel^T + h @ W2_root^T + b2).  K=128, N=40 (pad 48).
// Each wave owns 16 rows x 48 cols (3 accumulators) so the full 40-channel row
// is wave-resident; log-softmax via shfl_xor over the 16-lane halves.
__global__ __launch_bounds__(256) void gemm2_logsoftmax(
    const float* __restrict__ agg2, const float* __restrict__ h,
    const float* __restrict__ Wrel, const float* __restrict__ Wroot,
    const float* __restrict__ bias, float* __restrict__ out) {
  const int lane = threadIdx.x & 31;
  const int wave = threadIdx.x >> 5;
  const int mtile = blockIdx.x * 8 + wave;
  if (mtile >= N_NODES / 16) return;  // whole-wave exit: WMMA EXEC stays all-1s
  const int m_base = mtile * 16;
  const int l15 = lane & 15;
  const int kh = (lane >> 4) * 2;
  const int arow = m_base + l15;
  const int n2 = 32 + l15;
  const bool v2ok = (n2 < OUT_CH);
  const int n2c = v2ok ? n2 : 0;  // clamp address, value zeroed below

  const float* A0 = agg2 + (long long)arow * HID_CH + kh;
  const float* H0 = h + (long long)arow * HID_CH + kh;
  const float* Wr0 = Wrel + (long long)l15 * HID_CH + kh;
  const float* Wr1 = Wrel + (long long)(16 + l15) * HID_CH + kh;
  const float* Wr2 = Wrel + (long long)n2c * HID_CH + kh;
  const float* Wo0 = Wroot + (long long)l15 * HID_CH + kh;
  const float* Wo1 = Wroot + (long long)(16 + l15) * HID_CH + kh;
  const float* Wo2 = Wroot + (long long)n2c * HID_CH + kh;

  v8f c0 = {}, c1 = {}, c2 = {};
  for (int k = 0; k < HID_CH; k += 4) {
    v2f a = load_frag(A0 + k);
    v2f b0 = load_frag(Wr0 + k);
    v2f b1 = load_frag(Wr1 + k);
    v2f b2 = load_frag(Wr2 + k);
    if (!v2ok) { b2.x = 0.f; b2.y = 0.f; }
    c0 = __builtin_amdgcn_wmma_f32_16x16x4_f32(false, a, false, b0, (short)0,
                                               c0, false, false);
    c1 = __builtin_amdgcn_wmma_f32_16x16x4_f32(false, a, false, b1, (short)0,
                                               c1, false, false);
    c2 = __builtin_amdgcn_wmma_f32_16x16x4_f32(false, a, false, b2, (short)0,
                                               c2, false, false);
  }
  for (int k = 0; k < HID_CH; k += 4) {
    v2f a = load_frag(H0 + k);
    v2f b0 = load_frag(Wo0 + k);
    v2f b1 = load_frag(Wo1 + k);
    v2f b2 = load_frag(Wo2 + k);
    if (!v2ok) { b2.x = 0.f; b2.y = 0.f; }
    c0 = __builtin_amdgcn_wmma_f32_16x16x4_f32(false, a, false, b0, (short)0,
                                               c0, false, false);
    c1 = __builtin_amdgcn_wmma_f32_16x16x4_f32(false, a, false, b1, (short)0,
                                               c1, false, false);
    c2 = __builtin_amdgcn_wmma_f32_16x16x4_f32(false, a, false, b2, (short)0,
                                               c2, false, false);
  }

  const float bn0 = bias[l15];
  const float bn1 = bias[16 + l15];
  const float bn2 = v2ok ? bias[n2] : 0.0f;
  const int rbase = m_base + ((lane >> 4) << 3);
#pragma unroll
  for (int i = 0; i < 8; ++i) {
    float v0 = c0[i] + bn0;
    float v1 = c1[i] + bn1;
    float v2 = v2ok ? (c2[i] + bn2) : -INFINITY;
    // row-wise max over 40 channels: per-lane max of 3 tiles, then 16-lane tree
    float m = fmaxf(fmaxf(v0, v1), v2);
#pragma unroll
    for (int msk = 8; msk >= 1; msk >>= 1)
      m = fmaxf(m, __shfl_xor(m, msk, 32));  // masks <16 stay inside the half
    float s = __expf(v0 - m) + __expf(v1 - m) + (v2ok ? __expf(v2 - m) : 0.0f);
#pragma unroll
    for (int msk = 8; msk >= 1; msk >>= 1) s += __shfl_xor(s, msk, 32);
    const float lz = m + __logf(s);
    const long long ro = (long long)(rbase + i) * OUT_CH;
    out[ro + l15] = v0 - lz;
    out[ro + 16 + l15] = v1 - lz;
    if (v2ok) out[ro + n2] = v2 - lz;
  }
}

// ---------------------------------------------------------------------------
extern "C" void kernel_launch(void* const* d_in, const int* in_sizes, int n_in,
                              void* d_out, int out_size, void* d_ws,
                              size_t ws_size, hipStream_t stream) {
  (void)in_sizes; (void)n_in; (void)out_size; (void)ws_size;
  const float* x = (const float*)d_in[0];
  const int* ei = (const int*)d_in[1];  // [2, N_EDGES]
  const float* W1_rel = (const float*)d_in[2];
  const float* b1 = (const float*)d_in[3];
  const float* W1_root = (const float*)d_in[4];
  const float* W2_rel = (const float*)d_in[5];
  const float* b2 = (const float*)d_in[6];
  const float* W2_root = (const float*)d_in[7];
  float* out = (float*)d_out;

  // workspace layout: agg1 [N,64] | agg2 [N,128] | h [N,128]  (64 MB total)
  float* agg1 = (float*)d_ws;
  float* agg2 = agg1 + (size_t)N_NODES * IN_CH;
  float* hbuf = agg2 + (size_t)N_NODES * HID_CH;

  const int* srcv = ei;
  const int* dstv = ei + N_EDGES;

  // 1) zero agg1+agg2 (contiguous region)
  {
    long long n4 = (long long)N_NODES * (IN_CH + HID_CH) / 4;
    zero_f4<<<(int)((n4 + 255) / 256), 256, 0, stream>>>((float4*)agg1,
                                                         (int)n4);
  }
  // 2) agg1 = segment_sum(x[src], dst)
  {
    long long total = (long long)N_EDGES * (IN_CH / 4);
    scatter_add_kernel<4><<<(int)((total + 255) / 256), 256, 0, stream>>>(
        x, srcv, dstv, agg1);
  }
  // 3) h = relu(agg1 W1_rel^T + x W1_root^T + b1)
  gemm1_relu<<<N_NODES / 16, 256, 0, stream>>>(agg1, x, W1_rel, W1_root, b1,
                                               hbuf);
  // 4) agg2 = segment_sum(h[src], dst)
  {
    long long total = (long long)N_EDGES * (HID_CH / 4);
    scatter_add_kernel<5><<<(int)((total + 255) / 256), 256, 0, stream>>>(
        hbuf, srcv, dstv, agg2);
  }
  // 5) out = log_softmax(agg2 W2_rel^T + h W2_root^T + b2)
  gemm2_logsoftmax<<<(N_NODES / 16 + 7) / 8, 256, 0, stream>>>(
      agg2, hbuf, W2_rel, W2_root, b2, out);
}